// EMDLoss_16509854286170
// MI455X (gfx1250) — compile-verified
//
#include <hip/hip_runtime.h>

typedef __attribute__((ext_vector_type(2))) float v2f;
typedef __attribute__((ext_vector_type(8))) float v8f;

#define N_PTS     1024
#define N_TILES   64          // 1024 / 16
#define EPS_FINAL 0.0025f     // blur^p = 0.05^2
#define N_ANNEAL  12
#define N_EXTRA   5
#define N_ITERS   (N_ANNEAL + N_EXTRA)
#define LOG2E     1.4426950408889634f
#define LN2       0.6931471805599453f
#define NWAVES    32          // sinkhorn kernel uses 1024 threads

// Raw hardware transcendentals: v_exp_f32 computes 2^x, v_log_f32 computes log2(x)
__device__ __forceinline__ float fast_exp2(float x) { return __builtin_amdgcn_exp2f(x); }
__device__ __forceinline__ float fast_log2(float x) { return __builtin_amdgcn_logf(x); }

// ---------------------------------------------------------------------------
// Log2-domain Sinkhorn half-step. For each column j:
//   dual_col[j] = -(eps*ln2) * log2( sum_i 2^{ t2_ij } ),
//   t2_ij = log2e * ( logw + (dr_i - C_ij)/eps )
// One V_WMMA_F32_16X16X4_F32 per 16x16 tile (K-augmented, C = inline 0):
//   A row i = [x_i * log2e/eps (3), r_i],  r_i = logw2 + (dr_i - 0.5|x_i|^2)*log2e/eps
//   B col j = [y_j (3), 1]
//   per-column constant c_j = -0.5|y_j|^2 * log2e/eps folded into the LSE pass.
// Assumed f32 K=4 layouts (ISA 7.12.2): A/B lanes0-15 hold K=0(v0),1(v1);
// lanes16-31 hold K=2(v0),3(v1). C/D: lane = col, VGPR k = row k (+8 hi half).
// ---------------------------------------------------------------------------

// Streaming log2-sum-exp2 update; t excludes the per-column constant c.
__device__ __forceinline__ void lse2_update(float& m, float& s, const v8f& t, float c)
{
    const float tmax = fmaxf(fmaxf(fmaxf(t[0], t[1]), fmaxf(t[2], t[3])),
                             fmaxf(fmaxf(t[4], t[5]), fmaxf(t[6], t[7])));
    const float nm  = fmaxf(m, tmax + c);
    const float nmc = nm - c;
    // independent exps + tree sum: keeps the trans pipe full (no serial add chain)
    const float e0 = fast_exp2(t[0] - nmc);
    const float e1 = fast_exp2(t[1] - nmc);
    const float e2 = fast_exp2(t[2] - nmc);
    const float e3 = fast_exp2(t[3] - nmc);
    const float e4 = fast_exp2(t[4] - nmc);
    const float e5 = fast_exp2(t[5] - nmc);
    const float e6 = fast_exp2(t[6] - nmc);
    const float e7 = fast_exp2(t[7] - nmc);
    const float es = ((e0 + e1) + (e2 + e3)) + ((e4 + e5) + (e6 + e7));
    s = s * fast_exp2(m - nm) + es;
    m = nm;
}

// Precompute pre-swizzled WMMA A-rows for this phase into LDS staging:
// ARow2[2*i + 0] = {x*se, y*se},  ARow2[2*i + 1] = {z*se, r_i}
// so lane-half `hi` loads its v2f A operand with a single ds_load_b64.
__device__ __forceinline__ void build_rows(const float4* __restrict__ P,
                                           const float*  __restrict__ dual,
                                           float2*       __restrict__ ARow2,
                                           float inv_eps2, float logw2,
                                           int tid, int nthr)
{
    for (int i = tid; i < N_PTS; i += nthr) {
        const float4 p = P[i];
        ARow2[2 * i + 0] = make_float2(p.x * inv_eps2, p.y * inv_eps2);
        ARow2[2 * i + 1] = make_float2(p.z * inv_eps2,
                                       logw2 + (dual[i] - 0.5f * p.w) * inv_eps2);
    }
}

// One half-step: each wave owns column tiles {wave, wave+32}, jointly swept.
__device__ __forceinline__ void sinkhorn_phase(
    const float2* __restrict__ ARow2, // LDS pre-scaled, pre-swizzled row operands
    const float4* __restrict__ Pc,    // LDS col points {x,y,z,|p|^2}
    float*        __restrict__ dc,    // LDS output dual over cols
    float eps_ln2, float inv_eps2, int lane, int wave)
{
    const int  l16 = lane & 15;
    const bool hi  = (lane & 16) != 0;
    const int  asel = hi ? 1 : 0;
    const int  cb0 = wave << 4;                 // column tile 0 base
    const int  cb1 = (wave + NWAVES) << 4;      // column tile 1 base

    const float4 q0 = Pc[cb0 + l16];
    const float4 q1 = Pc[cb1 + l16];
    v2f B0, B1;
    B0[0] = hi ? q0.z : q0.x;  B0[1] = hi ? 1.0f : q0.y;
    B1[0] = hi ? q1.z : q1.x;  B1[1] = hi ? 1.0f : q1.y;
    const float c0 = -0.5f * q0.w * inv_eps2;
    const float c1 = -0.5f * q1.w * inv_eps2;

    float m0 = -__builtin_inff(), s0 = 0.0f;
    float m1 = -__builtin_inff(), s1 = 0.0f;

    const v8f vzero = {};   // C operand: inline 0 (cterm folded into LSE pass)

    for (int rt = 0; rt < N_TILES; ++rt) {
        const float2 ar = ARow2[(((rt << 4) + l16) << 1) + asel];
        v2f Am;
        Am[0] = ar.x;
        Am[1] = ar.y;

        v8f a0 = __builtin_amdgcn_wmma_f32_16x16x4_f32(false, Am, false, B0,
                                                       (short)0, vzero, false, false);
        v8f a1 = __builtin_amdgcn_wmma_f32_16x16x4_f32(false, Am, false, B1,
                                                       (short)0, vzero, false, false);
        lse2_update(m0, s0, a0, c0);
        lse2_update(m1, s1, a1, c1);
    }

    // merge lane L with L+16 (rows 0-7 with rows 8-15) and emit duals
    {
        const float mo = __shfl_xor(m0, 16, 32), so = __shfl_xor(s0, 16, 32);
        const float nm = fmaxf(m0, mo);
        const float st = s0 * fast_exp2(m0 - nm) + so * fast_exp2(mo - nm);
        if (!hi) dc[cb0 + l16] = -eps_ln2 * (nm + fast_log2(st));
    }
    {
        const float mo = __shfl_xor(m1, 16, 32), so = __shfl_xor(s1, 16, 32);
        const float nm = fmaxf(m1, mo);
        const float st = s1 * fast_exp2(m1 - nm) + so * fast_exp2(mo - nm);
        if (!hi) dc[cb1 + l16] = -eps_ln2 * (nm + fast_log2(st));
    }
}

__device__ __forceinline__ void load_cloud(const float* __restrict__ Pg,
                                           float4* __restrict__ Pl,
                                           int tid, int nthr)
{
    for (int i = tid; i < N_PTS; i += nthr) {
        const float a = Pg[3 * i], b = Pg[3 * i + 1], c = Pg[3 * i + 2];
        Pl[i] = make_float4(a, b, c, a * a + b * b + c * c);
    }
}

// ---------------------------------------------------------------------------
__global__ void init_ws_kernel(float* ws)
{
    if (threadIdx.x < 3) ws[threadIdx.x] = 0.0f;   // maxC slots (C>=0)
}

// exact max(C) per matrix type (xy/xx/yy), over all batches
__global__ __launch_bounds__(256) void cmax_kernel(
    const float* __restrict__ tmpl, const float* __restrict__ src,
    float* __restrict__ maxC)
{
    __shared__ float4 Xl[N_PTS], Yl[N_PTS];
    __shared__ float  red[8];

    const int tid = threadIdx.x;
    const int mat = blockIdx.x >> 3;
    const int b   = blockIdx.x & 7;
    const float* Xp = ((mat == 2) ? src : tmpl) + (size_t)b * N_PTS * 3;
    const float* Yp = ((mat == 0) ? src : (mat == 1) ? tmpl : src) + (size_t)b * N_PTS * 3;

    load_cloud(Xp, Xl, tid, 256);
    load_cloud(Yp, Yl, tid, 256);
    __syncthreads();

    float mx = 0.0f;
    for (int j = tid; j < N_PTS; j += 256) {
        const float4 q = Yl[j];
        for (int i = 0; i < N_PTS; ++i) {
            const float4 p = Xl[i];
            const float d0 = p.x - q.x, d1 = p.y - q.y, d2 = p.z - q.z;
            mx = fmaxf(mx, d0 * d0 + d1 * d1 + d2 * d2);
        }
    }
    mx *= 0.5f;
    #pragma unroll
    for (int off = 16; off > 0; off >>= 1) mx = fmaxf(mx, __shfl_xor(mx, off, 32));
    if ((tid & 31) == 0) red[tid >> 5] = mx;
    __syncthreads();
    if (tid == 0) {
        float t = red[0];
        for (int w = 1; w < 8; ++w) t = fmaxf(t, red[w]);
        // C >= 0 -> float ordering == int-bit ordering
        atomicMax((int*)&maxC[mat], __float_as_int(t));
    }
}

// persistent Sinkhorn: one workgroup per (matrix, batch), 32 waves
__global__ __launch_bounds__(1024) void sinkhorn_kernel(
    const float* __restrict__ tmpl, const float* __restrict__ src,
    const float* __restrict__ maxC, float* __restrict__ ot_out)
{
    __shared__ float4 Xl[N_PTS], Yl[N_PTS];
    __shared__ float2 ARow2[2 * N_PTS];
    __shared__ float  fv[N_PTS], gv[N_PTS];
    __shared__ float  red[NWAVES];

    const int tid  = threadIdx.x;
    const int nthr = blockDim.x;
    const int lane = tid & 31;
    const int wave = tid >> 5;
    const int mat  = blockIdx.x >> 3;
    const int b    = blockIdx.x & 7;

    const float* Xp = ((mat == 2) ? src : tmpl) + (size_t)b * N_PTS * 3;
    const float* Yp = ((mat == 0) ? src : (mat == 1) ? tmpl : src) + (size_t)b * N_PTS * 3;

    load_cloud(Xp, Xl, tid, nthr);
    load_cloud(Yp, Yl, tid, nthr);
    for (int i = tid; i < N_PTS; i += nthr) { fv[i] = 0.0f; gv[i] = 0.0f; }
    __syncthreads();

    const float logw2 = -10.0f;                              // log2(1/1024)
    const float eps0  = fmaxf(maxC[mat], 2.0f * EPS_FINAL);  // ~diameter^2 start
    const float lr    = logf(EPS_FINAL / eps0) / (float)(N_ANNEAL - 1);

    for (int it = 0; it < N_ITERS; ++it) {
        const float eps      = (it < N_ANNEAL) ? eps0 * expf(lr * (float)it) : EPS_FINAL;
        const float inv_eps2 = LOG2E / eps;
        const float eps_ln2  = eps * LN2;

        // g_j = -eps * LSE_i( log a + (f_i - C_ij)/eps )
        build_rows(Xl, fv, ARow2, inv_eps2, logw2, tid, nthr);
        __syncthreads();
        sinkhorn_phase(ARow2, Yl, gv, eps_ln2, inv_eps2, lane, wave);
        __syncthreads();
        // f_i = -eps * LSE_j( log b + (g_j - C_ij)/eps )
        build_rows(Yl, gv, ARow2, inv_eps2, logw2, tid, nthr);
        __syncthreads();
        sinkhorn_phase(ARow2, Xl, fv, eps_ln2, inv_eps2, lane, wave);
        __syncthreads();
    }

    // OT value = mean(f) + mean(g)
    float acc = 0.0f;
    for (int i = tid; i < N_PTS; i += nthr) acc += fv[i] + gv[i];
    #pragma unroll
    for (int off = 16; off > 0; off >>= 1) acc += __shfl_xor(acc, off, 32);
    if (lane == 0) red[wave] = acc;
    __syncthreads();
    if (tid == 0) {
        float t = 0.0f;
        for (int w = 0; w < NWAVES; ++w) t += red[w];
        ot_out[blockIdx.x] = t / (float)N_PTS;
    }
}

__global__ void combine_kernel(const float* __restrict__ ot, float* __restrict__ out)
{
    if (threadIdx.x == 0) {
        float acc = 0.0f;
        for (int b = 0; b < 8; ++b)
            acc += ot[b] - 0.5f * (ot[8 + b] + ot[16 + b]);   // debias
        out[0] = acc / (8.0f * (float)N_PTS);                 // (/N).mean over batch
    }
}

// ---------------------------------------------------------------------------
extern "C" void kernel_launch(void* const* d_in, const int* in_sizes, int n_in,
                              void* d_out, int out_size, void* d_ws, size_t ws_size,
                              hipStream_t stream)
{
    (void)in_sizes; (void)n_in; (void)out_size; (void)ws_size;
    const float* tmpl = (const float*)d_in[0];   // [8,1024,3] f32
    const float* src  = (const float*)d_in[1];   // [8,1024,3] f32
    float* ws   = (float*)d_ws;
    float* maxC = ws;         // 3 floats
    float* ot   = ws + 8;     // 24 floats (mat*8 + b)

    init_ws_kernel<<<1, 32, 0, stream>>>(ws);
    cmax_kernel<<<24, 256, 0, stream>>>(tmpl, src, maxC);
    sinkhorn_kernel<<<24, 1024, 0, stream>>>(tmpl, src, maxC, ot);
    combine_kernel<<<1, 32, 0, stream>>>(ot, (float*)d_out);
}